// ScaledDotProductAttentionStatic_71184787963921
// MI455X (gfx1250) — compile-verified
//
#include <hip/hip_runtime.h>

typedef float v2f __attribute__((ext_vector_type(2)));
typedef float v8f __attribute__((ext_vector_type(8)));

// Problem constants (reference: B,H,S,D = 2,8,2048,64; TEMPERATURE = 8)
constexpr int kB  = 2;
constexpr int kH  = 8;
constexpr int kS  = 2048;
constexpr int kD  = 64;
constexpr int kQT = 16;            // query rows per block
constexpr int kKT = kS / 16;       // 128 key tiles
constexpr int kStride = 2052;      // padded LDS row stride (floats), 2052 % 64 == 4

__global__ __launch_bounds__(256)
void attn_static_fused(const float* __restrict__ q,
                       const float* __restrict__ k,
                       const float* __restrict__ v,
                       const float* __restrict__ cov,
                       const float* __restrict__ lamw,
                       const int*   __restrict__ mask,
                       float* __restrict__ out,
                       float* __restrict__ attn)
{
    extern __shared__ float smem[];
    float* sc  = smem;                      // [16][kStride] logits -> probs
    float* qs  = smem + kQT * kStride;      // [16][64] scaled Q tile
    float* psc = qs + kQT * kD;             // [4][256] PV partial combine

    const int tid   = threadIdx.x;
    const int lane  = tid & 31;
    const int wave  = tid >> 5;
    const int row16 = lane & 15;
    const int hi    = (lane >> 4) & 1;

    const int qt = blockIdx.x % (kS / kQT);
    const int h  = (blockIdx.x / (kS / kQT)) % kH;
    const int b  =  blockIdx.x / ((kS / kQT) * kH);

    const float lam    = lamw[h];
    const float wqk    = 1.0f / (lam + 1.0f);
    const float wcov   = lam  / (lam + 1.0f);
    const float qscale = wqk * 0.125f;      // fold w_qk and 1/TEMPERATURE into Q

    const size_t bh      = (size_t)b * kH + h;
    const size_t headoff = bh * (size_t)kS * kD;
    const size_t bhS     = bh * (size_t)kS;

    // ---- Phase 0: load + scale Q tile into LDS (one float4 per thread) ----
    {
        const float4* qg = (const float4*)(q + headoff + (size_t)qt * kQT * kD);
        float4 t4 = qg[tid];
        t4.x *= qscale; t4.y *= qscale; t4.z *= qscale; t4.w *= qscale;
        ((float4*)qs)[tid] = t4;
    }
    __syncthreads();

    // A-fragments for Q: 16 slices of K=4 (ISA layout: V0 = K{0,2}, V1 = K{1,3})
    v2f aQ[16];
    #pragma unroll
    for (int s = 0; s < 16; ++s) {
        const float* p = qs + row16 * kD + 4 * s + 2 * hi;
        aQ[s][0] = p[0];
        aQ[s][1] = p[1];
    }

    // ---- Phase 1: scores = (Q*wqk/T) K^T, + wcov*cov, mask, into LDS ----
    const float* kg = k + headoff;
    for (int kt = wave; kt < kKT; kt += 8) {
        const int    kcol = kt * 16 + row16;
        const float* kp   = kg + (size_t)kcol * kD;

        // Issue cov/mask loads first (unconditional, straight-line) so they
        // batch into one wait and hide under the WMMA chain below.
        const size_t rbase = ((size_t)b * kS + qt * kQT + 8 * hi) * (size_t)kS + kcol;
        float cvv[8];
        int   mkv[8];
        #pragma unroll
        for (int r = 0; r < 8; ++r) cvv[r] = cov[rbase + (size_t)r * kS];
        #pragma unroll
        for (int r = 0; r < 8; ++r) mkv[r] = mask[rbase + (size_t)r * kS];

        // Prefetch next key tile owned by this wave (global_prefetch_b8).
        if (kt + 8 < kKT) {
            __builtin_prefetch(kp + (size_t)128 * kD, 0, 0);
            __builtin_prefetch(cov + rbase + 128, 0, 0);
        }

        v8f acc = {};
        #pragma unroll
        for (int s = 0; s < 16; ++s) {
            v2f bK;
            const float* bp = kp + 4 * s + 2 * hi;
            bK[0] = bp[0]; bK[1] = bp[1];
            acc = __builtin_amdgcn_wmma_f32_16x16x4_f32(
                false, aQ[s], false, bK, (short)0, acc, false, false);
        }

        #pragma unroll
        for (int r = 0; r < 8; ++r) {
            const float t = fmaf(wcov, cvv[r], acc[r]);   // branchless select below
            sc[(r + 8 * hi) * kStride + kcol] = (mkv[r] != 0) ? t : -1.0e9f;
        }
    }
    __syncthreads();

    // ---- Phase 2: row softmax (2 rows per wave), write attn to global ----
    for (int rr = 0; rr < 2; ++rr) {
        const int row = wave * 2 + rr;
        float* srow = sc + row * kStride;

        float m = -3.4e38f;
        for (int c = lane; c < kS; c += 32) m = fmaxf(m, srow[c]);
        #pragma unroll
        for (int off = 16; off > 0; off >>= 1) m = fmaxf(m, __shfl_xor(m, off, 32));

        float sum = 0.0f;
        for (int c = lane; c < kS; c += 32) {
            const float e = __expf(srow[c] - m);
            srow[c] = e;
            sum += e;
        }
        #pragma unroll
        for (int off = 16; off > 0; off >>= 1) sum += __shfl_xor(sum, off, 32);

        const float inv = 1.0f / sum;
        float4* g4 = (float4*)(attn + (bhS + qt * kQT + row) * (size_t)kS);
        float4* s4 = (float4*)srow;
        for (int c = lane; c < kS / 4; c += 32) {
            float4 t = s4[c];
            t.x *= inv; t.y *= inv; t.z *= inv; t.w *= inv;
            s4[c] = t;   // normalized probs kept in LDS for PV phase
            g4[c] = t;   // streamed to global attn (dominant HBM traffic)
        }
    }
    __syncthreads();

    // ---- Phase 3: out = P @ V  (4 D-tiles x 2 K-halves across 8 waves) ----
    const int ct    = wave & 3;       // which 16-wide column tile of D
    const int kh    = wave >> 2;      // K half: [0,1024) or [1024,2048)
    const int dbase = ct * 16;
    const float* vg = v + headoff + dbase + row16;

    v8f acc = {};
    const int k0 = kh * (kS / 2), k1 = k0 + (kS / 2);
    #pragma unroll 4
    for (int kb = k0; kb < k1; kb += 4) {
        const int t0 = kb + 2 * hi;
        v2f aP, bV;
        const float* ap = sc + row16 * kStride + t0;
        aP[0] = ap[0]; aP[1] = ap[1];
        bV[0] = vg[(size_t)t0 * kD];
        bV[1] = vg[(size_t)(t0 + 1) * kD];
        acc = __builtin_amdgcn_wmma_f32_16x16x4_f32(
            false, aP, false, bV, (short)0, acc, false, false);
    }

    if (kh == 1) {
        #pragma unroll
        for (int r = 0; r < 8; ++r) psc[ct * 256 + r * 32 + lane] = acc[r];
    }
    __syncthreads();
    if (kh == 0) {
        float* og = out + (bhS + qt * kQT) * (size_t)kD;
        #pragma unroll
        for (int r = 0; r < 8; ++r) {
            const float val = acc[r] + psc[ct * 256 + r * 32 + lane];
            const int   qrl = r + 8 * hi;
            og[(size_t)qrl * kD + dbase + row16] = val;
        }
    }
}

extern "C" void kernel_launch(void* const* d_in, const int* in_sizes, int n_in,
                              void* d_out, int out_size, void* d_ws, size_t ws_size,
                              hipStream_t stream) {
    (void)in_sizes; (void)n_in; (void)d_ws; (void)ws_size; (void)out_size;
    const float* q    = (const float*)d_in[0];
    const float* k    = (const float*)d_in[1];
    const float* v    = (const float*)d_in[2];
    const float* cov  = (const float*)d_in[3];
    const float* lamw = (const float*)d_in[4];
    const int*   mask = (const int*)  d_in[5];

    float* out  = (float*)d_out;                              // [B,H,S,D]
    float* attn = out + (size_t)kB * kH * kS * kD;            // [B,H,S,S]

    const int    blocks = kB * kH * (kS / kQT);               // 2048
    const size_t shmem  = (size_t)(kQT * kStride + kQT * kD + 4 * 256) * sizeof(float);

    attn_static_fused<<<blocks, 256, shmem, stream>>>(q, k, v, cov, lamw, mask, out, attn);
}